// LengthRegulator_20323785244726
// MI455X (gfx1250) — compile-verified
//
#include <hip/hip_runtime.h>
#include <hip/hip_bf16.h>
#include <stdint.h>

// Problem constants from the reference.
#define B_      16
#define L_      512
#define H_      384      // floats per row = 96 float4
#define MAXLEN_ 4096
#define TWO_L   1024     // 2*L entries in the repeat/cumsum arrays

// ---------------------------------------------------------------------------
// Kernel 1: per-batch repeat computation + inclusive scan (cumsum) in LDS.
// One block per batch, 1024 threads (= 2L), wave32 -> 32 waves.
// ---------------------------------------------------------------------------
__global__ __launch_bounds__(TWO_L) void lr_scan_kernel(
    const float* __restrict__ pd, const float* __restrict__ sd,
    const int* __restrict__ src_lens, int* __restrict__ cum_ws,
    int* __restrict__ tgt_ws, float* __restrict__ out_tail) {
  const int b = blockIdx.x;
  const int i = threadIdx.x;          // 0..1023
  const int l = i >> 1;
  const bool is_sil = (i & 1) != 0;
  __shared__ int s[TWO_L];

  const int sl = src_lens[b];
  const bool valid = l < sl;
  // Reference: round(where(valid, dur, 0)) then clamp. jnp.round == half-to-even
  // == rintf under default rounding mode. Invalid phone slots still yield 1.
  const float d = valid ? (is_sil ? sd[b * L_ + l] : pd[b * L_ + l]) : 0.0f;
  const int r = (int)rintf(d);
  const int rep = is_sil ? (r > 0 ? r : 0) : (r > 1 ? r : 1);
  s[i] = rep;
  __syncthreads();

  // Hillis-Steele inclusive scan over 1024 entries (10 steps).
  for (int off = 1; off < TWO_L; off <<= 1) {
    const int v   = s[i];
    const int add = (i >= off) ? s[i - off] : 0;
    __syncthreads();
    s[i] = v + add;
    __syncthreads();
  }

  cum_ws[b * TWO_L + i] = s[i];
  if (i == TWO_L - 1) {
    tgt_ws[b]   = s[i];
    out_tail[b] = (float)s[i];   // tgt_len, second tuple output
  }
}

// ---------------------------------------------------------------------------
// Kernel 2: expand rows. Each wave (32 lanes) owns one output row per
// iteration: wave-uniform binary search in the LDS-resident cumsum, then a
// fully coalesced 1536B row copy as 3x float4 per lane.
// The 4KB cumsum table is staged via CDNA5 async global->LDS DMA.
// ---------------------------------------------------------------------------
#define ROWS_PB  32     // output rows per block
#define WAVES_PB 8      // 256 threads / 32

__global__ __launch_bounds__(256) void lr_gather_kernel(
    const float* __restrict__ x, const int* __restrict__ cum_ws,
    const int* __restrict__ tgt_ws, float* __restrict__ out) {
  const int b    = blockIdx.y;
  const int row0 = blockIdx.x * ROWS_PB;
  const int tid  = threadIdx.x;
  const int lane = tid & 31;
  const int wave = tid >> 5;
  __shared__ int s_cum[TWO_L];

  // Async DMA: 256 lanes x 16B = 4KB cumsum table, global -> LDS directly
  // (no VGPR round-trip), tracked by ASYNCcnt.
  {
    unsigned lds = (unsigned)(uintptr_t)(&s_cum[0]) + (unsigned)(tid * 16);
    unsigned long long ga =
        (unsigned long long)(uintptr_t)(cum_ws + (size_t)b * TWO_L) +
        (unsigned long long)(tid * 16);
    asm volatile("global_load_async_to_lds_b128 %0, %1, off"
                 :: "v"(lds), "v"(ga) : "memory");
    asm volatile("s_wait_asynccnt 0" ::: "memory");
  }
  __syncthreads();

  const int tgt = tgt_ws[b];
  const float* xb = x + (size_t)b * L_ * H_;
  float* ob = out + ((size_t)b * MAXLEN_ + row0) * (size_t)H_;

  for (int r = wave; r < ROWS_PB; r += WAVES_PB) {
    const int t = row0 + r;                 // wave-uniform
    float4 v0, v1, v2;
    if (t < tgt) {
      // searchsorted(cum, t, side='right'): first idx with cum[idx] > t.
      int lo = 0, hi = TWO_L;
      #pragma unroll
      for (int step = 0; step < 10; ++step) {
        const int mid = (lo + hi) >> 1;
        if (s_cum[mid] <= t) lo = mid + 1; else hi = mid;
      }
      int seg = lo < (TWO_L - 1) ? lo : (TWO_L - 1);
      // Odd segment = silence -> source row 0 (matches reference exactly).
      const int src = (seg & 1) ? 0 : (seg >> 1);
      const float4* xr = (const float4*)(xb + (size_t)src * H_);
      v0 = xr[lane];
      v1 = xr[lane + 32];
      v2 = xr[lane + 64];
    } else {
      v0 = make_float4(0.f, 0.f, 0.f, 0.f);
      v1 = v0; v2 = v0;
    }
    float4* orow = (float4*)(ob + (size_t)r * H_);
    orow[lane]      = v0;
    orow[lane + 32] = v1;
    orow[lane + 64] = v2;
  }
}

// ---------------------------------------------------------------------------
extern "C" void kernel_launch(void* const* d_in, const int* in_sizes, int n_in,
                              void* d_out, int out_size, void* d_ws, size_t ws_size,
                              hipStream_t stream) {
  (void)in_sizes; (void)n_in; (void)out_size; (void)ws_size;

  const float* x        = (const float*)d_in[0];
  const float* pd       = (const float*)d_in[1];
  const float* sd       = (const float*)d_in[2];
  const int*   src_lens = (const int*)d_in[3];
  // d_in[4] = max_len scalar on device; value is the compile-time 4096.

  float* out      = (float*)d_out;
  float* out_tail = out + (size_t)B_ * MAXLEN_ * H_;   // tgt_len slots

  int* cum_ws = (int*)d_ws;                // 16 * 1024 ints = 64 KB
  int* tgt_ws = cum_ws + B_ * TWO_L;       // 16 ints

  lr_scan_kernel<<<dim3(B_), dim3(TWO_L), 0, stream>>>(
      pd, sd, src_lens, cum_ws, tgt_ws, out_tail);

  lr_gather_kernel<<<dim3(MAXLEN_ / ROWS_PB, B_), dim3(256), 0, stream>>>(
      x, cum_ws, tgt_ws, out);
}